// DecorrLoss_781684048735
// MI455X (gfx1250) — compile-verified
//
#include <hip/hip_runtime.h>

typedef __attribute__((ext_vector_type(2))) float v2f;
typedef __attribute__((ext_vector_type(8))) float v8f;

#define D       512
#define NSAMP   32768
#define SK      32                   // split-K slices
#define KCHUNK  (NSAMP / SK)         // 1024 samples per slice
#define BK      16                   // K-panel depth (4 WMMA k-steps)
#define BM      128                  // block tile is BM x BM
#define NTILES  10                   // 4x4 lower-triangle tile pairs
#define LROW    (2 * BM + 32)        // 288: K-interleaved row; +32 pad => 288 % 64 == 32
                                     // -> the two lane-halves hit complementary bank sets

// ---------------------------------------------------------------------------
// Zero workspace (gram + 2 scalar accumulators)
// ---------------------------------------------------------------------------
__global__ void zero_kernel(float* __restrict__ p, int n) {
    int idx = blockIdx.x * blockDim.x + threadIdx.x;
    if (idx < n) p[idx] = 0.0f;
}

// ---------------------------------------------------------------------------
// Per-sample s2 / s4 reduction -> loss partials
// accum[0] += s2^2 - s4 ; accum[1] += s4 - 2*s2   (+d folded in at finalize)
// ---------------------------------------------------------------------------
__global__ __launch_bounds__(256)
void loss_kernel(const float* __restrict__ x, float* __restrict__ accum) {
    const int lane = threadIdx.x & 31;
    const int wave = threadIdx.x >> 5;
    const long row = (long)blockIdx.x * 8 + wave;          // one wave per sample
    const float* __restrict__ xr = x + row * D;

    float s2 = 0.0f, s4 = 0.0f;
    #pragma unroll 4
    for (int c = lane; c < D; c += 32) {
        float v  = xr[c];
        float v2 = v * v;
        s2 += v2;
        s4 += v2 * v2;
    }
    #pragma unroll
    for (int off = 16; off > 0; off >>= 1) {
        s2 += __shfl_xor(s2, off, 32);
        s4 += __shfl_xor(s4, off, 32);
    }

    __shared__ float red[2][8];
    if (lane == 0) {
        red[0][wave] = s2 * s2 - s4;
        red[1][wave] = s4 - 2.0f * s2;
    }
    __syncthreads();
    if (threadIdx.x == 0) {
        float c = 0.0f, w = 0.0f;
        #pragma unroll
        for (int i = 0; i < 8; ++i) { c += red[0][i]; w += red[1][i]; }
        atomicAdd(&accum[0], c);
        atomicAdd(&accum[1], w);
    }
}

// ---------------------------------------------------------------------------
// gram += X^T X : 128x128 lower-triangle tiles, fp32 WMMA 16x16x4.
// LDS panels stored K-pair-interleaved so every WMMA fragment is one
// aligned ds_load_b64 (no repacking movs). Double-buffered LDS, one
// barrier per K-panel, register prefetch of the next panel.
// Each wave32 computes a 32x64 patch (2x4 accumulator tiles).
// ---------------------------------------------------------------------------
__global__ __launch_bounds__(256)
void gram_wmma_kernel(const float* __restrict__ x, float* __restrict__ gram) {
    __shared__ float As[2][8 * LROW];
    __shared__ float Bs[2][8 * LROW];

    const int tid     = threadIdx.x;
    const int tile_id = blockIdx.x % NTILES;
    const int kslice  = blockIdx.x / NTILES;

    // decode lower-triangle pair: tile_id = tm*(tm+1)/2 + tn, tn <= tm
    int tm = 0, tn = tile_id;
    while (tn > tm) { ++tm; tn -= tm; }
    const int i0 = tm * BM;            // row-feature block
    const int j0 = tn * BM;            // col-feature block

    const int wave = tid >> 5;
    const int lane = tid & 31;
    const int wr   = wave >> 1;        // 0..3 -> +32 rows
    const int wc   = wave & 1;         // 0..1 -> +64 cols
    const int half = lane >> 4;        // K-half select for f32 fragments
    const int lr   = lane & 15;

    v8f acc[2][4] = {};

    // panel fill mapping: kpair = wave (0..7), 4 consecutive features/thread
    const int kp = tid >> 5;
    const int fg = (tid & 31) * 4;

    const long nbase = (long)kslice * KCHUNK;
    const float* __restrict__ pa = x + (nbase + 2 * kp) * D + i0 + fg;
    const float* __restrict__ pb = x + (nbase + 2 * kp) * D + j0 + fg;

    float4 a0 = *(const float4*)(pa);
    float4 a1 = *(const float4*)(pa + D);
    float4 b0 = *(const float4*)(pb);
    float4 b1 = *(const float4*)(pb + D);

    const int ITERS = KCHUNK / BK;     // 64
    for (int it = 0; it < ITERS; ++it) {
        const int buf = it & 1;

        // K-interleaved store: the LDS pair for kpair kp and feature f holds that feature from samples n0+2kp and n0+2kp+1
        float2* __restrict__ sa = (float2*)&As[buf][kp * LROW + fg * 2];
        float2* __restrict__ sb = (float2*)&Bs[buf][kp * LROW + fg * 2];
        sa[0] = make_float2(a0.x, a1.x);
        sa[1] = make_float2(a0.y, a1.y);
        sa[2] = make_float2(a0.z, a1.z);
        sa[3] = make_float2(a0.w, a1.w);
        sb[0] = make_float2(b0.x, b1.x);
        sb[1] = make_float2(b0.y, b1.y);
        sb[2] = make_float2(b0.z, b1.z);
        sb[3] = make_float2(b0.w, b1.w);
        __syncthreads();

        if (it + 1 < ITERS) {          // prefetch next panel (uniform branch)
            pa += (long)BK * D;
            pb += (long)BK * D;
            a0 = *(const float4*)(pa);
            a1 = *(const float4*)(pa + D);
            b0 = *(const float4*)(pb);
            b1 = *(const float4*)(pb + D);
        }

        #pragma unroll
        for (int k4 = 0; k4 < 4; ++k4) {
            // lanes 0-15 take kpair 2*k4 (K = 4k4,4k4+1); lanes 16-31 kpair 2*k4+1
            const int krow = (2 * k4 + half) * LROW;

            v2f af[2], bf[4];
            #pragma unroll
            for (int s = 0; s < 2; ++s)
                af[s] = *(const v2f*)&As[buf][krow + (wr * 32 + s * 16 + lr) * 2];
            #pragma unroll
            for (int t = 0; t < 4; ++t)
                bf[t] = *(const v2f*)&Bs[buf][krow + (wc * 64 + t * 16 + lr) * 2];

            #pragma unroll
            for (int s = 0; s < 2; ++s)
                #pragma unroll
                for (int t = 0; t < 4; ++t)
                    acc[s][t] = __builtin_amdgcn_wmma_f32_16x16x4_f32(
                        false, af[s], false, bf[t], (short)0, acc[s][t], false, false);
        }
    }

    // C/D layout: VGPR r, lanes 0-15 -> (M=r, N=lane); lanes 16-31 -> (M=r+8, N=lane-16)
    #pragma unroll
    for (int s = 0; s < 2; ++s)
        #pragma unroll
        for (int t = 0; t < 4; ++t)
            #pragma unroll
            for (int r = 0; r < 8; ++r) {
                const int gi = i0 + wr * 32 + s * 16 + r + 8 * half;
                const int gj = j0 + wc * 64 + t * 16 + lr;
                atomicAdd(&gram[gi * D + gj], acc[s][t][r]);
            }
}

// ---------------------------------------------------------------------------
// Finalize: grad + losses
// ---------------------------------------------------------------------------
__global__ __launch_bounds__(256)
void finalize_kernel(const float* __restrict__ gram,
                     const float* __restrict__ accum,
                     const float* __restrict__ kappa_p,
                     float* __restrict__ out) {
    const int idx = blockIdx.x * blockDim.x + threadIdx.x;
    const float kappa = *kappa_p;
    const float invN  = 1.0f / (float)NSAMP;

    if (idx < D * D) {
        const int i = idx / D;
        const int j = idx % D;
        // gram stored only where tile(i) >= tile(j) (128-wide tiles); mirror otherwise
        int a = i, b = j;
        if ((i / BM) < (j / BM)) { a = j; b = i; }
        const float g = gram[a * D + b] * invN;
        out[idx] = (i == j) ? kappa * (g - 1.0f) : (1.0f - kappa) * g;
    }
    if (idx == 0) {
        const float inv_d2 = 1.0f / ((float)D * (float)D);
        out[D * D]     = accum[0] * invN * inv_d2;                  // correlation
        out[D * D + 1] = (accum[1] * invN + (float)D) * inv_d2;     // whitening
    }
}

// ---------------------------------------------------------------------------
extern "C" void kernel_launch(void* const* d_in, const int* in_sizes, int n_in,
                              void* d_out, int out_size, void* d_ws, size_t ws_size,
                              hipStream_t stream) {
    const float* x     = (const float*)d_in[0];
    const float* kappa = (const float*)d_in[1];
    float* out   = (float*)d_out;
    float* gram  = (float*)d_ws;           // D*D floats
    float* accum = gram + D * D;           // 2 floats

    const int nz = D * D + 2;
    zero_kernel<<<(nz + 255) / 256, 256, 0, stream>>>(gram, nz);
    loss_kernel<<<NSAMP / 8, 256, 0, stream>>>(x, accum);
    gram_wmma_kernel<<<NTILES * SK, 256, 0, stream>>>(x, gram);
    finalize_kernel<<<(D * D + 255) / 256, 256, 0, stream>>>(gram, accum, kappa, out);
}